// Mseloss_75419625718249
// MI455X (gfx1250) — compile-verified
//
#include <hip/hip_runtime.h>
#include <stdint.h>

// Problem constants (fixed per reference)
#define T_FRAMES 1024
#define BATCH    2
#define NVOCAB   10000
#define LGL      30           // glosses per sample
#define SSTATES  (2*LGL + 1)  // 61 CTC-extended states (fits in a 64-bit mask)

#define TOT_BLOCKS  1000      // 1000*256 threads -> exactly 20 float4 per thread
#define TOT_THREADS 256
#define MASK_BLOCKS ((BATCH*T_FRAMES)/256)   // = 8

// workspace layout (bytes) — every word read is rewritten each call
#define PMASK_OFF 0                     // float pmask[MASK_BLOCKS]
#define PTOT_OFF  64                    // float ptot[TOT_BLOCKS]

typedef __attribute__((ext_vector_type(2))) float v2f;
typedef __attribute__((ext_vector_type(8))) float v8f;

// ---------------------------------------------------------------------------
// Kernel 1: masked sum of squares. The CTC reachability recurrence is
// monotone (state bits only accumulate), so the 61-bit lattice mask reaches
// its fixpoint within <= 60 steps. Each block redundantly runs the forward
// and backward scans for its batch sample for just 64 steps in LDS, then
// every thread evaluates sel[t] = rf(min(t,63)) & rb(min(T-1-t,63)) directly.
// One thread per (b,t); blank (v=0) masked iff any even lattice state active;
// each distinct active label vocab counted once (dedup over 30 glosses).
// ---------------------------------------------------------------------------
__global__ void masked_ssq_kernel(const float* __restrict__ logits,
                                  const int* __restrict__ label,
                                  float* __restrict__ pmask)
{
    __shared__ int lab[LGL];
    __shared__ unsigned long long rfArr[64];
    __shared__ unsigned long long rbArr[64];
    __shared__ float red[256];

    const int tid = threadIdx.x;
    const int b   = blockIdx.x >> 2;                 // blocks 0-3 -> b=0, 4-7 -> b=1
    const int t   = ((blockIdx.x & 3) << 8) + tid;   // t in [0, 1024)

    if (tid < LGL) lab[tid] = label[b * LGL + tid];
    __syncthreads();

    const unsigned long long SMASK = (1ULL << SSTATES) - 1ULL;

    if (tid == 0) {
        // forward skip-2 allowed at odd state j=2k+1 iff k==0 || lab[k]!=lab[k-1]
        unsigned long long a2f = 0ULL;
        for (int k = 0; k < LGL; ++k)
            if (k == 0 || lab[k] != lab[k - 1]) a2f |= 1ULL << (2 * k + 1);
        unsigned long long c = 0x3ULL;               // states 0,1 at time 0
        rfArr[0] = c;
        for (int s = 1; s < 64; ++s) {               // saturates by s<=59
            c = (c | (c << 1) | ((c << 2) & a2f)) & SMASK;
            rfArr[s] = c;
        }
    } else if (tid == 1) {
        // backward skip-2 allowed at odd j=2k+1 iff k==L-1 || lab[k]!=lab[k+1]
        unsigned long long a2b = 0ULL;
        for (int k = 0; k < LGL; ++k)
            if (k == LGL - 1 || lab[k] != lab[k + 1]) a2b |= 1ULL << (2 * k + 1);
        unsigned long long c = 0x3ULL << (SSTATES - 2); // states S-1,S-2 at T-1
        rbArr[0] = c;                                   // distance-from-end = 0
        for (int s = 1; s < 64; ++s) {
            c = c | (c >> 1) | ((c >> 2) & a2b);
            rbArr[s] = c;
        }
    }
    __syncthreads();

    const int d = T_FRAMES - 1 - t;                  // distance from end
    const unsigned long long rf = rfArr[t < 64 ? t : 63];
    const unsigned long long rb = rbArr[d < 64 ? d : 63];
    const unsigned long long s  = rf & rb;
    const unsigned long long EVEN = 0x5555555555555555ULL & SMASK;

    const float* row = logits + ((size_t)t * BATCH + b) * NVOCAB;
    float acc = 0.0f;
    if (s & EVEN) { float x = row[0]; acc += x * x; }          // blank
    for (int k = 0; k < LGL; ++k) {
        if (s & (1ULL << (2 * k + 1))) {
            const int v = lab[k];
            bool dup = false;
            for (int k2 = 0; k2 < k; ++k2)
                if ((s & (1ULL << (2 * k2 + 1))) && lab[k2] == v) { dup = true; break; }
            if (!dup) { float x = row[v]; acc += x * x; }
        }
    }

    red[tid] = acc;
    __syncthreads();
    for (int off = 128; off > 0; off >>= 1) {
        if (tid < off) red[tid] += red[tid + off];
        __syncthreads();
    }
    if (tid == 0) pmask[blockIdx.x] = red[0];
}

// ---------------------------------------------------------------------------
// Kernel 2: bandwidth-bound total sum of squares over all 20.48M logits.
// Exact-fit grid (no tail, no exec-mask overhead): 256,000 threads x 20
// float4 loads each, unrolled with 4 independent accumulators so multiple
// b128 loads stay in flight (L2-resident working set rewards MLP).
// Wave32 reduction via V_WMMA_F32_16X16X4_F32 (A = lane accumulators,
// B = ones): D[m,n] = acc_m + acc_{m+16}; summing a lane's 8 D VGPRs gives
// the half-wave sum, one ds_swizzle SWAPX16 completes the wave sum.
// ---------------------------------------------------------------------------
__global__ void total_ssq_kernel(const float* __restrict__ logits,
                                 float* __restrict__ ptot)
{
    const float4* p = (const float4*)logits;
    const size_t stride = (size_t)TOT_BLOCKS * TOT_THREADS;   // 256,000
    size_t idx = (size_t)blockIdx.x * TOT_THREADS + threadIdx.x;

    float a0 = 0.f, a1 = 0.f, a2 = 0.f, a3 = 0.f;
#pragma unroll
    for (int k = 0; k < 5; ++k) {                 // 5 x 4 = 20 float4 / thread
        float4 x0 = p[idx];
        float4 x1 = p[idx + stride];
        float4 x2 = p[idx + 2 * stride];
        float4 x3 = p[idx + 3 * stride];
        idx += 4 * stride;
        a0 = fmaf(x0.x, x0.x, a0); a0 = fmaf(x0.y, x0.y, a0);
        a0 = fmaf(x0.z, x0.z, a0); a0 = fmaf(x0.w, x0.w, a0);
        a1 = fmaf(x1.x, x1.x, a1); a1 = fmaf(x1.y, x1.y, a1);
        a1 = fmaf(x1.z, x1.z, a1); a1 = fmaf(x1.w, x1.w, a1);
        a2 = fmaf(x2.x, x2.x, a2); a2 = fmaf(x2.y, x2.y, a2);
        a2 = fmaf(x2.z, x2.z, a2); a2 = fmaf(x2.w, x2.w, a2);
        a3 = fmaf(x3.x, x3.x, a3); a3 = fmaf(x3.y, x3.y, a3);
        a3 = fmaf(x3.z, x3.z, a3); a3 = fmaf(x3.w, x3.w, a3);
    }
    float acc = (a0 + a1) + (a2 + a3);

    // wave32 reduction via FP32 WMMA (exec all-1s: exact grid, no divergence)
    v2f a;  a.x = acc;  a.y = 0.0f;   // A[m,0]=acc_m, A[m,2]=acc_{m+16}, rest 0
    v2f bm; bm.x = 1.0f; bm.y = 1.0f; // B = ones(4x16)
    v8f c = {};
    c = __builtin_amdgcn_wmma_f32_16x16x4_f32(false, a, false, bm,
                                              (short)0, c, false, false);
    float s = c[0] + c[1] + c[2] + c[3] + c[4] + c[5] + c[6] + c[7];
    // swap halves of the wave: ds_swizzle group-of-32, xor=0x10, and=0x1f
    float other = __int_as_float(
        __builtin_amdgcn_ds_swizzle(__float_as_int(s), 0x401F));
    float wsum = s + other;

    __shared__ float wred[TOT_THREADS / 32];
    const int lane = threadIdx.x & 31;
    const int wv   = threadIdx.x >> 5;
    if (lane == 0) wred[wv] = wsum;
    __syncthreads();
    if (threadIdx.x == 0) {
        float bsum = 0.0f;
        for (int w = 0; w < TOT_THREADS / 32; ++w) bsum += wred[w];
        ptot[blockIdx.x] = bsum;
    }
}

// ---------------------------------------------------------------------------
// Kernel 3: deterministic finalize in double: out = 0.5 * (total - masked).
// ---------------------------------------------------------------------------
__global__ void finalize_kernel(const float* __restrict__ pmask,
                                const float* __restrict__ ptot,
                                float* __restrict__ out)
{
    __shared__ double red[256];
    const int tid = threadIdx.x;
    double acc = 0.0;
    for (int i = tid; i < TOT_BLOCKS;  i += 256) acc += (double)ptot[i];
    for (int i = tid; i < MASK_BLOCKS; i += 256) acc -= (double)pmask[i];
    red[tid] = acc;
    __syncthreads();
    for (int off = 128; off > 0; off >>= 1) {
        if (tid < off) red[tid] += red[tid + off];
        __syncthreads();
    }
    if (tid == 0) out[0] = (float)(0.5 * red[0]);
}

extern "C" void kernel_launch(void* const* d_in, const int* in_sizes, int n_in,
                              void* d_out, int out_size, void* d_ws, size_t ws_size,
                              hipStream_t stream) {
    const float* logits = (const float*)d_in[0];   // (T, B, N) f32
    const int*   label  = (const int*)d_in[2];     // (B*L,) i32   (lens fixed)

    uint8_t* ws = (uint8_t*)d_ws;
    float* pmask = (float*)(ws + PMASK_OFF);
    float* ptot  = (float*)(ws + PTOT_OFF);
    float* out   = (float*)d_out;

    masked_ssq_kernel<<<MASK_BLOCKS, 256, 0, stream>>>(logits, label, pmask);
    total_ssq_kernel<<<TOT_BLOCKS, TOT_THREADS, 0, stream>>>(logits, ptot);
    finalize_kernel<<<1, 256, 0, stream>>>(pmask, ptot, out);
}